// CRY_88871463288930
// MI455X (gfx1250) — compile-verified
//
#include <hip/hip_runtime.h>

// CRY gate, DIM=2, N=24, C=0, T=1, J=1, K=2, B=2.
// Per real/imag plane (2^25 floats): control bit = float-index bit 24,
// target bit = bit 23. control=0 half is identity copy; control=1 half mixes
// the two target blocks with [[cos,-sin],[-sin,cos]] of theta/2.
// Pure stream: 512 MB traffic -> ~22 us floor at 23.3 TB/s. WMMA is
// intentionally unused (0.03 flop/byte); the win is b128 NT vmem with trig
// hidden behind in-flight loads.

typedef float float4v __attribute__((ext_vector_type(4)));

static constexpr unsigned COPY_UNITS = 1u << 22; // float4 units in control=0 half
static constexpr unsigned ROT_UNITS  = 1u << 21; // float4 pair-units in control=1 half
static constexpr unsigned IMAG_OFS4  = 1u << 23; // float4 offset of imag plane in d_out

// Branch-free sincos, valid to ~1e-7 abs for |x| <~ 1e3 (here |x| <~ 10).
// 3-term Cody-Waite pi/2 reduction + minimax polys on [-pi/4, pi/4].
__device__ __forceinline__ void fast_sincos(float x, float& s_out, float& c_out) {
    float n  = rintf(x * 0.63661977236758134f);   // x * 2/pi
    int   q  = (int)n;
    float r  = fmaf(n, -1.57079601287841796875f, x);  // pi/2 hi
    r        = fmaf(n, -3.139164786504813e-7f,  r);   // pi/2 mid
    r        = fmaf(n, -5.390302529957765e-15f, r);   // pi/2 lo
    float r2 = r * r;
    // sin(r) ~= r + r^3*(S1 + r^2*(S2 + r^2*S3))
    float ps = fmaf(r * r2,
                    fmaf(r2, fmaf(r2, -1.95152959e-4f, 8.33216087e-3f),
                         -1.66666547e-1f),
                    r);
    // cos(r) ~= 1 + r^2*(C1 + r^2*(C2 + r^2*C3))
    float pc = fmaf(r2,
                    fmaf(r2, fmaf(r2, -1.388731625e-3f, 4.16666418e-2f), -0.5f),
                    1.0f);
    bool  swap = (q & 1) != 0;
    float sv = swap ? pc : ps;
    float cv = swap ? ps : pc;
    s_out = (q & 2)       ? -sv : sv;   // quadrants 2,3 negate sin
    c_out = ((q + 1) & 2) ? -cv : cv;   // quadrants 1,2 negate cos
}

__global__ __launch_bounds__(256) void cry_stream_kernel(
    const float4v* __restrict__ xr,   // x_real, 2^23 float4s
    const float4v* __restrict__ xi,   // x_imag, 2^23 float4s
    const float*   __restrict__ angle,
    float4v*       __restrict__ out)  // [real plane | imag plane], 2^24 float4s
{
    unsigned i = blockIdx.x * blockDim.x + threadIdx.x;

    if (i < COPY_UNITS) {
        // control = 0: identity copy of both planes
        float4v r = __builtin_nontemporal_load(&xr[i]);
        float4v m = __builtin_nontemporal_load(&xi[i]);
        __builtin_nontemporal_store(r, &out[i]);
        __builtin_nontemporal_store(m, &out[i + IMAG_OFS4]);
    } else {
        // control = 1: rotate the (target=0, target=1) pair, both planes.
        // Grid covers exactly COPY_UNITS+ROT_UNITS threads, so no upper check.
        unsigned j  = i - COPY_UNITS;
        unsigned v0 = COPY_UNITS + j;        // target = 0 block
        unsigned v1 = v0 + ROT_UNITS;        // target = 1 block

        // Issue all four NT loads first; trig overlaps the memory latency.
        float4v r0 = __builtin_nontemporal_load(&xr[v0]);
        float4v r1 = __builtin_nontemporal_load(&xr[v1]);
        float4v m0 = __builtin_nontemporal_load(&xi[v0]);
        float4v m1 = __builtin_nontemporal_load(&xi[v1]);

        float cv, sv;
        fast_sincos(0.5f * angle[0], sv, cv);

        float4v or0 = cv * r0 - sv * r1;     // out[jm] = cos*x0 - sin*x1
        float4v or1 = cv * r1 - sv * r0;     // out[km] = -sin*x0 + cos*x1
        float4v om0 = cv * m0 - sv * m1;
        float4v om1 = cv * m1 - sv * m0;

        __builtin_nontemporal_store(or0, &out[v0]);
        __builtin_nontemporal_store(or1, &out[v1]);
        __builtin_nontemporal_store(om0, &out[v0 + IMAG_OFS4]);
        __builtin_nontemporal_store(om1, &out[v1 + IMAG_OFS4]);
    }
}

extern "C" void kernel_launch(void* const* d_in, const int* in_sizes, int n_in,
                              void* d_out, int out_size, void* d_ws, size_t ws_size,
                              hipStream_t stream) {
    (void)in_sizes; (void)n_in; (void)out_size; (void)d_ws; (void)ws_size;

    const float4v* xr    = (const float4v*)d_in[0];
    const float4v* xi    = (const float4v*)d_in[1];
    const float*   angle = (const float*)d_in[2];
    float4v*       out   = (float4v*)d_out;

    const unsigned total   = COPY_UNITS + ROT_UNITS;   // 6,291,456 threads
    const unsigned threads = 256;                       // 8 wave32 per block
    const unsigned blocks  = total / threads;           // 24,576 (exact)

    cry_stream_kernel<<<blocks, threads, 0, stream>>>(xr, xi, angle, out);
}